// DepthCueExtractor_49392123904764
// MI455X (gfx1250) — compile-verified
//
#include <hip/hip_runtime.h>
#include <hip/hip_bf16.h>

typedef __attribute__((ext_vector_type(2))) float v2f;
typedef __attribute__((ext_vector_type(8))) float v8f;

#define B_ 4
#define A_ 9
#define H_ 256
#define W_ 256
#define C_ 9
#define F_ 64

// ---------------------------------------------------------------------------
// Kernel 1: hv[b,w,f] = sum_h f_maps[b,h,w,f]
// f_maps is [B,H,W,F]; each thread owns one (b, w*F+f) and strides over H.
// Fully coalesced across threads (contiguous w*F+f), 67 MB read total.
// The 1/H of the mean cancels in the later max-normalization, so plain sum.
// ---------------------------------------------------------------------------
__global__ void k_colsum(const float* __restrict__ f_maps, float* __restrict__ hv) {
    int tid = blockIdx.x * blockDim.x + threadIdx.x;      // 0 .. B*W*F-1
    int b   = tid / (W_ * F_);
    int wf  = tid % (W_ * F_);
    const float* p = f_maps + (size_t)b * H_ * W_ * F_ + wf;
    float s = 0.0f;
#pragma unroll 8
    for (int h = 0; h < H_; ++h) s += p[(size_t)h * (W_ * F_)];
    hv[tid] = s;
}

// ---------------------------------------------------------------------------
// Kernel 2: per (b,f): mx = max_w hv[b,w,f]; hv[b,w,f] /= mx   (in place)
// Only B*F = 256 independent columns; tiny.
// ---------------------------------------------------------------------------
__global__ void k_norm(float* __restrict__ hv) {
    int tid = threadIdx.x;                                 // 0 .. B*F-1
    int b = tid / F_, f = tid % F_;
    float* p = hv + (size_t)b * W_ * F_ + f;
    float mx = -INFINITY;
    for (int w = 0; w < W_; ++w) mx = fmaxf(mx, p[(size_t)w * F_]);
    float inv = 1.0f / mx;
    for (int w = 0; w < W_; ++w) p[(size_t)w * F_] *= inv;
}

// ---------------------------------------------------------------------------
// Kernel 3 (fused): one block per (b,h).
//   m[w]   = (1/81) * sum_{a,c} lfi[b,a,h,w,c]          -> LDS (W floats)
//   row[f] = hvn[b,h,f]                                  -> LDS (F floats)
//   out[b,h,w,f] = m[w] * row[f]  via V_WMMA_F32_16X16X4_F32 rank-1 tiles
// 256 threads = 8 wave32s; 16x4 = 64 output tiles of 16x16; 8 tiles/wave.
// Wave wv covers w-tiles {2wv, 2wv+1} x f-tiles {0..3}: preload the 2 m
// values and 4 row values per lane with unconditional ds_loads (valid index
// for all 32 lanes), then zero the upper half-wave with a register select
// (v_cndmask) -- no divergent EXEC save/restore around the WMMAs.
// A tile: m column in K=0 (lanes 0-15, VGPR0), zeros elsewhere.
// B tile: hvn row in K=0 (lanes 0-15, VGPR0), zeros elsewhere.
// D layout: VGPR r -> row r (lanes 0-15) / row r+8 (lanes 16-31), col = lane&15.
// ---------------------------------------------------------------------------
__global__ void k_outer(const float* __restrict__ lfi,
                        const float* __restrict__ hvn,
                        float* __restrict__ out) {
    __shared__ float m_s[W_];
    __shared__ float f_s[F_];

    int bh = blockIdx.x;               // b*H + h
    int b  = bh / H_;
    int h  = bh % H_;
    int t  = threadIdx.x;              // 0..255, one thread per w

    // ---- reduce lfi over (a, c) for w = t ----
    {
        const float* base = lfi + (((size_t)b * A_) * H_ + h) * ((size_t)W_ * C_)
                                + (size_t)t * C_;
        __builtin_prefetch(base, 0, 0);                    // global_prefetch_b8
        float s = 0.0f;
#pragma unroll
        for (int a = 0; a < A_; ++a) {
            const float* pa = base + (size_t)a * H_ * W_ * C_;
#pragma unroll
            for (int c = 0; c < C_; ++c) s += pa[c];
        }
        m_s[t] = s * (1.0f / (float)(A_ * C_));
    }
    if (t < F_) f_s[t] = hvn[((size_t)b * W_ + h) * F_ + t];
    __syncthreads();

    const int  lane = t & 31;
    const int  wv   = t >> 5;          // wave id 0..7
    const int  col  = lane & 15;
    const bool lo   = lane < 16;
    float* obase = out + (size_t)bh * (W_ * F_);

    // Preload this wave's operand slices (uniform ds_loads, all lanes).
    float mv0 = m_s[(2 * wv + 0) * 16 + col];
    float mv1 = m_s[(2 * wv + 1) * 16 + col];
    float fv0 = f_s[0 * 16 + col];
    float fv1 = f_s[1 * 16 + col];
    float fv2 = f_s[2 * 16 + col];
    float fv3 = f_s[3 * 16 + col];
    // Zero the upper half-wave in registers (K=2/K=3 slots of A and B).
    mv0 = lo ? mv0 : 0.0f;  mv1 = lo ? mv1 : 0.0f;
    fv0 = lo ? fv0 : 0.0f;  fv1 = lo ? fv1 : 0.0f;
    fv2 = lo ? fv2 : 0.0f;  fv3 = lo ? fv3 : 0.0f;
    const float fsel[4] = {fv0, fv1, fv2, fv3};
    const int rb = lo ? 0 : 8;

#pragma unroll
    for (int i = 0; i < 8; ++i) {
        int wt = 2 * wv + (i >> 2);    // w-tile
        int ft = i & 3;                // f-tile

        v2f a;  a.x  = (i < 4) ? mv0 : mv1;  a.y  = 0.0f;
        v2f bm; bm.x = fsel[ft];             bm.y = 0.0f;
        v8f c = {};
        // D = A(16x4) * B(4x16) + 0   (only K=0 populated -> rank-1 product)
        v8f d = __builtin_amdgcn_wmma_f32_16x16x4_f32(
            /*neg_a=*/false, a, /*neg_b=*/false, bm,
            /*c_mod=*/(short)0, c, /*reuse_a=*/false, /*reuse_b=*/false);

#pragma unroll
        for (int r = 0; r < 8; ++r) {
            obase[(size_t)(wt * 16 + rb + r) * F_ + ft * 16 + col] = d[r];
        }
    }
}

// ---------------------------------------------------------------------------
extern "C" void kernel_launch(void* const* d_in, const int* in_sizes, int n_in,
                              void* d_out, int out_size, void* d_ws, size_t ws_size,
                              hipStream_t stream) {
    const float* lfi    = (const float*)d_in[0];   // [B,A,H,W,C] fp32
    const float* f_maps = (const float*)d_in[1];   // [B,H,W,F]   fp32
    float* out = (float*)d_out;                    // [B,H,W,F]   fp32
    float* hv  = (float*)d_ws;                     // B*W*F floats = 256 KB scratch

    k_colsum<<<(B_ * W_ * F_) / 256, 256, 0, stream>>>(f_maps, hv);
    k_norm<<<1, B_ * F_, 0, stream>>>(hv);
    k_outer<<<B_ * H_, 256, 0, stream>>>(lfi, hv, out);
}